// DBSCANModel_59622736003665
// MI455X (gfx1250) — compile-verified
//
#include <hip/hip_runtime.h>
#include <stdint.h>

// ---------------- problem constants ----------------
#define N_PTS   12288
#define DIMS    64
#define W_WORDS (N_PTS / 32)   // 384 adjacency words per row
#define SENT    N_PTS
#define EPS2    110.25f        // eps = 10.5, compare d2 <= eps^2
#define MIN_SAMP 5

typedef float v2f __attribute__((ext_vector_type(2)));
typedef float v8f __attribute__((ext_vector_type(8)));

// ---------------- K0: per-row squared norms ----------------
__global__ void dbscan_sq_kernel(const float* __restrict__ X, float* __restrict__ sq) {
    int i = blockIdx.x * blockDim.x + threadIdx.x;
    if (i >= N_PTS) return;
    const float4* row = (const float4*)(X + (size_t)i * DIMS);
    float s = 0.0f;
#pragma unroll
    for (int t = 0; t < DIMS / 4; t++) {
        float4 v = row[t];
        s += v.x * v.x + v.y * v.y + v.z * v.z + v.w * v.w;
    }
    sq[i] = s;
}

// ---------------- K1: WMMA f32 distance tiles -> adjacency bit matrix ---------
// Each wave computes a 16-row x 64-col tile with FOUR 16x16x4-f32 WMMA
// accumulators (5 fragment loads per 4 WMMAs). 8 waves per block share the
// same row tile i0 so A-fragments are reused 8x through WGP$/L2.
// A(16x4) layout: lane&15 = M, lanes>=16 hold K+2..K+3; B(4x16) symmetric.
// C layout: VGPR v -> row (v or v+8), lane&15 -> col. ballot_w32 packs
// adjacency bits; each row emits 2 words, stored as one b64.
__global__ __launch_bounds__(256)
void dbscan_adj_kernel(const float* __restrict__ X, const float* __restrict__ sq,
                       uint32_t* __restrict__ adj) {
    const int lane = threadIdx.x & 31;
    const int wid  = threadIdx.x >> 5;
    const int i0 = blockIdx.y * 16;                    // row tile (shared by block)
    const int j0 = (blockIdx.x * 8 + wid) * 64;        // 64-col tile per wave
    const int m    = lane & 15;
    const int koff = (lane >> 4) * 2;                  // lanes 16..31 hold K+2,K+3

    const float* arow = X + (size_t)(i0 + m) * DIMS + koff;
    const float* brow0 = X + (size_t)(j0 + m)      * DIMS + koff;
    const float* brow1 = X + (size_t)(j0 + 16 + m) * DIMS + koff;
    const float* brow2 = X + (size_t)(j0 + 32 + m) * DIMS + koff;
    const float* brow3 = X + (size_t)(j0 + 48 + m) * DIMS + koff;

    v8f acc0 = {}, acc1 = {}, acc2 = {}, acc3 = {};
#pragma unroll
    for (int k = 0; k < DIMS; k += 4) {
        v2f a  = *(const v2f*)(arow  + k);
        v2f b0 = *(const v2f*)(brow0 + k);
        v2f b1 = *(const v2f*)(brow1 + k);
        v2f b2 = *(const v2f*)(brow2 + k);
        v2f b3 = *(const v2f*)(brow3 + k);
        acc0 = __builtin_amdgcn_wmma_f32_16x16x4_f32(false, a, false, b0,
                                                     (short)0, acc0, false, false);
        acc1 = __builtin_amdgcn_wmma_f32_16x16x4_f32(false, a, false, b1,
                                                     (short)0, acc1, false, false);
        acc2 = __builtin_amdgcn_wmma_f32_16x16x4_f32(false, a, false, b2,
                                                     (short)0, acc2, false, false);
        acc3 = __builtin_amdgcn_wmma_f32_16x16x4_f32(false, a, false, b3,
                                                     (short)0, acc3, false, false);
    }

    const float sqc0 = sq[j0 + m];
    const float sqc1 = sq[j0 + 16 + m];
    const float sqc2 = sq[j0 + 32 + m];
    const float sqc3 = sq[j0 + 48 + m];
    const int   roff = (lane >> 4) * 8;  // lanes 16..31 hold rows M+8

    uint32_t myw0 = 0, myw1 = 0;
#pragma unroll
    for (int v = 0; v < 8; v++) {
        const float sqr = sq[i0 + roff + v];
        bool p0 = (sqr + sqc0 - 2.0f * acc0[v]) <= EPS2;
        bool p1 = (sqr + sqc1 - 2.0f * acc1[v]) <= EPS2;
        bool p2 = (sqr + sqc2 - 2.0f * acc2[v]) <= EPS2;
        bool p3 = (sqr + sqc3 - 2.0f * acc3[v]) <= EPS2;
        uint32_t m0 = __builtin_amdgcn_ballot_w32(p0);
        uint32_t m1 = __builtin_amdgcn_ballot_w32(p1);
        uint32_t m2 = __builtin_amdgcn_ballot_w32(p2);
        uint32_t m3 = __builtin_amdgcn_ballot_w32(p3);
        // low 16 ballot bits = row i0+v, high 16 = row i0+8+v
        uint32_t wA0 = (m0 & 0xffffu) | ((m1 & 0xffffu) << 16); // row i0+v,   cols j0..j0+31
        uint32_t wA1 = (m2 & 0xffffu) | ((m3 & 0xffffu) << 16); // row i0+v,   cols j0+32..j0+63
        uint32_t wB0 = (m0 >> 16)     | (m1 & 0xffff0000u);     // row i0+8+v, cols j0..j0+31
        uint32_t wB1 = (m2 >> 16)     | (m3 & 0xffff0000u);     // row i0+8+v, cols j0+32..j0+63
        if (lane == v)     { myw0 = wA0; myw1 = wA1; }
        if (lane == v + 8) { myw0 = wB0; myw1 = wB1; }
    }
    if (lane < 16) {
        uint32_t* dst = adj + (size_t)(i0 + lane) * W_WORDS + (j0 >> 5);
        // (j0>>5) is even -> 8B aligned -> single global_store_b64
        uint2 pair; pair.x = myw0; pair.y = myw1;
        *(uint2*)dst = pair;
    }
}

// ---------------- K2: degree / core flag / initial labels ----------------
__global__ void dbscan_core_kernel(const uint32_t* __restrict__ adj,
                                   int* __restrict__ core, int* __restrict__ lab) {
    int i = blockIdx.x * blockDim.x + threadIdx.x;
    if (i >= N_PTS) return;
    const uint32_t* row = adj + (size_t)i * W_WORDS;
    int deg = 0;
    for (int t = 0; t < W_WORDS; t++) deg += __popc(row[t]);
    int c = (deg >= MIN_SAMP) ? 1 : 0;
    core[i] = c;
    lab[i]  = c ? i : SENT;
}

// ---------------- K3: core bitmask (1 bit per point) ----------------
__global__ void dbscan_coremask_kernel(const int* __restrict__ core,
                                       uint32_t* __restrict__ coremask) {
    int t = blockIdx.x * blockDim.x + threadIdx.x;
    if (t >= W_WORDS) return;
    uint32_t w = 0;
    for (int b = 0; b < 32; b++) w |= (core[t * 32 + b] ? 1u : 0u) << b;
    coremask[t] = w;
}

// ---------------- K4: min-label relaxation over core-core edges -------------
__global__ void dbscan_relax_kernel(const uint32_t* __restrict__ adj,
                                    const uint32_t* __restrict__ coremask,
                                    const int* __restrict__ core,
                                    int* __restrict__ lab) {
    int wave = (blockIdx.x * blockDim.x + threadIdx.x) >> 5;
    int lane = threadIdx.x & 31;
    if (wave >= N_PTS) return;
    if (!core[wave]) return;   // wave-uniform
    const uint32_t* row = adj + (size_t)wave * W_WORDS;
    int m = SENT;
    for (int t = lane; t < W_WORDS; t += 32) {
        uint32_t w = row[t] & coremask[t];
        int base = t * 32;
        while (w) {
            int b = __builtin_ctz(w);
            w &= (w - 1);
            int l = lab[base + b];
            m = (l < m) ? l : m;
        }
    }
#pragma unroll
    for (int off = 16; off > 0; off >>= 1) {
        int o = __shfl_xor(m, off, 32);
        m = (o < m) ? o : m;
    }
    if (lane == 0) atomicMin(&lab[wave], m);
}

// ---------------- K5: pointer jumping (path compression) ----------------
__global__ void dbscan_jump_kernel(int* __restrict__ lab) {
    int i = blockIdx.x * blockDim.x + threadIdx.x;
    if (i >= N_PTS) return;
    int l = lab[i];
    if (l < SENT) {
        int l2 = lab[l];
        if (l2 < l) lab[i] = l2;
    }
}

// ---------------- K6: border-point representative ----------------
__global__ void dbscan_border_kernel(const uint32_t* __restrict__ adj,
                                     const uint32_t* __restrict__ coremask,
                                     const int* __restrict__ core,
                                     const int* __restrict__ lab,
                                     int* __restrict__ rep) {
    int wave = (blockIdx.x * blockDim.x + threadIdx.x) >> 5;
    int lane = threadIdx.x & 31;
    if (wave >= N_PTS) return;
    const uint32_t* row = adj + (size_t)wave * W_WORDS;
    int m = SENT;
    for (int t = lane; t < W_WORDS; t += 32) {
        uint32_t w = row[t] & coremask[t];
        int base = t * 32;
        while (w) {
            int b = __builtin_ctz(w);
            w &= (w - 1);
            int l = lab[base + b];
            m = (l < m) ? l : m;
        }
    }
#pragma unroll
    for (int off = 16; off > 0; off >>= 1) {
        int o = __shfl_xor(m, off, 32);
        m = (o < m) ? o : m;
    }
    if (lane == 0) rep[wave] = core[wave] ? lab[wave] : m;
}

// ---------------- K7: inclusive prefix sum of is_rep (single block) ---------
__global__ void dbscan_scan_kernel(const int* __restrict__ core,
                                   const int* __restrict__ lab,
                                   int* __restrict__ prefix) {
    __shared__ int sums[256];
    const int tid = threadIdx.x;
    const int chunk = N_PTS / 256;   // 48
    const int base = tid * chunk;
    int s = 0;
    for (int k = 0; k < chunk; k++) {
        int i = base + k;
        s += (core[i] && lab[i] == i) ? 1 : 0;
    }
    sums[tid] = s;
    __syncthreads();
    // Hillis-Steele inclusive scan over 256 partials
    for (int off = 1; off < 256; off <<= 1) {
        int v = (tid >= off) ? sums[tid - off] : 0;
        __syncthreads();
        sums[tid] += v;
        __syncthreads();
    }
    int run = (tid == 0) ? 0 : sums[tid - 1];   // exclusive for this chunk
    for (int k = 0; k < chunk; k++) {
        int i = base + k;
        run += (core[i] && lab[i] == i) ? 1 : 0;
        prefix[i] = run;                        // inclusive cumsum
    }
}

// ---------------- K8: final labels ----------------
__global__ void dbscan_final_kernel(const int* __restrict__ rep,
                                    const int* __restrict__ prefix,
                                    int* __restrict__ out) {
    int i = blockIdx.x * blockDim.x + threadIdx.x;
    if (i >= N_PTS) return;
    int r = rep[i];
    out[i] = (r >= SENT) ? -1 : (prefix[r] - 1);
}

// ---------------- launcher ----------------
extern "C" void kernel_launch(void* const* d_in, const int* in_sizes, int n_in,
                              void* d_out, int out_size, void* d_ws, size_t ws_size,
                              hipStream_t stream) {
    const float* X = (const float*)d_in[0];
    int* out = (int*)d_out;   // reference output dtype is int32

    // workspace carve-up (256B aligned)
    size_t off = 0;
    auto carve = [&](size_t bytes) -> char* {
        char* p = (char*)d_ws + off;
        off = (off + bytes + 255) & ~(size_t)255;
        return p;
    };
    uint32_t* adj      = (uint32_t*)carve((size_t)N_PTS * W_WORDS * 4); // 18.9 MB
    float*    sq       = (float*)   carve((size_t)N_PTS * 4);
    int*      core     = (int*)     carve((size_t)N_PTS * 4);
    uint32_t* coremask = (uint32_t*)carve((size_t)W_WORDS * 4);
    int*      lab      = (int*)     carve((size_t)N_PTS * 4);
    int*      rep      = (int*)     carve((size_t)N_PTS * 4);
    int*      prefix   = (int*)     carve((size_t)N_PTS * 4);
    if (off > ws_size) return;   // insufficient scratch; avoid OOB

    const int TPB = 256;
    const int ROW_BLOCKS  = N_PTS / TPB;                 // 48
    const int WAVE_BLOCKS = N_PTS / (TPB / 32);          // 1536 (wave per row)

    dbscan_sq_kernel<<<ROW_BLOCKS, TPB, 0, stream>>>(X, sq);

    // 8 waves/block, each wave a 16x64 tile: grid (12288/512, 12288/16)
    dim3 adjGrid(N_PTS / (64 * 8), N_PTS / 16);          // (24, 768)
    dbscan_adj_kernel<<<adjGrid, 256, 0, stream>>>(X, sq, adj);

    dbscan_core_kernel<<<ROW_BLOCKS, TPB, 0, stream>>>(adj, core, lab);
    dbscan_coremask_kernel<<<(W_WORDS + TPB - 1) / TPB, TPB, 0, stream>>>(core, coremask);

    // Shiloach-Vishkin style: relax + pointer-jump, fixed log-bounded rounds
    for (int it = 0; it < 16; it++) {
        dbscan_relax_kernel<<<WAVE_BLOCKS, TPB, 0, stream>>>(adj, coremask, core, lab);
        dbscan_jump_kernel <<<ROW_BLOCKS,  TPB, 0, stream>>>(lab);
        dbscan_jump_kernel <<<ROW_BLOCKS,  TPB, 0, stream>>>(lab);
    }

    dbscan_border_kernel<<<WAVE_BLOCKS, TPB, 0, stream>>>(adj, coremask, core, lab, rep);
    dbscan_scan_kernel  <<<1, 256, 0, stream>>>(core, lab, prefix);
    dbscan_final_kernel <<<ROW_BLOCKS, TPB, 0, stream>>>(rep, prefix, out);
}